// orgCTree_47124381172234
// MI455X (gfx1250) — compile-verified
//
#include <hip/hip_runtime.h>
#include <hip/hip_bf16.h>
#include <cstdint>
#include <cstddef>

#define HH 2048
#define WW 2048
#define CC 14
#define TILE 32
#define HALO 34
#define HPIX (HALO * HALO) /* 1156 */
#define NTX 64
#define NTY 64
#define NTILES (NTX * NTY)

typedef __attribute__((ext_vector_type(2))) float v2f;
typedef __attribute__((ext_vector_type(8))) float v8f;
typedef __attribute__((ext_vector_type(4))) unsigned int v4u;
typedef __attribute__((ext_vector_type(8))) int v8i;
typedef __attribute__((ext_vector_type(4))) int v4i;

#if defined(__HIP_DEVICE_COMPILE__) && defined(__gfx1250__) && __has_builtin(__builtin_amdgcn_tensor_load_to_lds)
#define HAVE_TDM 1
#else
#define HAVE_TDM 0
#endif

__device__ __forceinline__ void tdm_wait(int n) {
#if __has_builtin(__builtin_amdgcn_s_wait_tensorcnt)
  if (n == 0) __builtin_amdgcn_s_wait_tensorcnt(0);
  else        __builtin_amdgcn_s_wait_tensorcnt(1);
#else
  if (n == 0) asm volatile("s_wait_tensorcnt 0x0" ::: "memory");
  else        asm volatile("s_wait_tensorcnt 0x1" ::: "memory");
#endif
}

#if HAVE_TDM
// Issue one 34x34 f32 tile DMA (global -> LDS), 2D tensor descriptor.
// Interior tiles only: every element is in-range, contiguous LDS dest.
__device__ __forceinline__ void tdm_load_tile(const float* gsrc, float* ldst) {
  uint64_t ga = (uint64_t)(uintptr_t)gsrc;
  uint32_t la = (uint32_t)(uintptr_t)ldst; // low 32 bits of generic LDS addr = LDS byte offset
  v4u g0 = { 1u,                                   // count=1, is_restore=0, gather off
             la,                                   // lds_addr
             (uint32_t)ga,                         // global_addr[31:0]
             (uint32_t)((ga >> 32) & 0x01FFFFFFu) | (2u << 30) }; // addr[56:32] | type=2
  v8i g1 = { (int)0x00020000u,                                   // data_size=4B, no flags
             (int)(((uint32_t)WW & 0xFFFFu) << 16),              // tensor_dim0 lo16 @ [31:16]
             (int)(((uint32_t)WW >> 16) | (((uint32_t)HH & 0xFFFFu) << 16)), // dim0 hi | dim1 lo
             (int)(((uint32_t)HH >> 16) | ((uint32_t)HALO << 16)),           // dim1 hi | tile_dim0
             (int)((uint32_t)HALO),                               // tile_dim1=34, tile_dim2=0
             (int)WW,                                             // tensor_dim0_stride lo32
             0, 0 };
  v4i gz4 = { 0, 0, 0, 0 };
  v8i gz8 = { 0, 0, 0, 0, 0, 0, 0, 0 };
  // amdgpu-toolchain (clang-23) 6-arg form: (g0, g1, g2, g3, g4, cpol)
  __builtin_amdgcn_tensor_load_to_lds(g0, g1, gz4, gz4, gz8, 0);
}
#endif

__global__ void k_init(unsigned int* __restrict__ hdr) {
  int i = threadIdx.x;
  if (i < CC) {
    hdr[i]          = 0u;           // count of maxima
    hdr[CC + i]     = 0x7F800000u;  // min bits (+inf)
    hdr[2 * CC + i] = 0u;           // top1 bits
    hdr[3 * CC + i] = 0u;           // top2 bits
  }
}

__global__ __launch_bounds__(256) void k_tile(const float* __restrict__ x,
                                              unsigned int* __restrict__ hdr,
                                              float* __restrict__ psum) {
  __shared__ float sm_m[HPIX];
  __shared__ float sm_r[HPIX];
  __shared__ float sm_x[2][HPIX];
  __shared__ float sm_p[HPIX];
  __shared__ float rS[256], rM[256], rA[256], rB[256];
  __shared__ int   rI[256];

  const int tid = threadIdx.x;
  const int tx = blockIdx.x, ty = blockIdx.y;
  const int gx0 = tx * TILE - 1, gy0 = ty * TILE - 1;
#if HAVE_TDM
  const bool edge = (tx == 0) | (ty == 0) | (tx == NTX - 1) | (ty == NTY - 1);
  if (!edge && tid < 32) { // wave 0 pre-issues channel 0 tile; overlaps with phase 1
    tdm_load_tile(x + (size_t)gy0 * WW + gx0, &sm_x[0][0]);
  }
#else
  const bool edge = true;
#endif

  // Phase 1: per-pixel softmax stats (max, 1/sum-exp) over channels for the halo tile.
  for (int hp = tid; hp < HPIX; hp += 256) {
    int hy = hp / HALO, hx = hp - hy * HALO;
    int gy = gy0 + hy, gx = gx0 + hx;
    float m = 0.f, r = 0.f;
    if ((unsigned)gy < (unsigned)HH && (unsigned)gx < (unsigned)WW) {
      const float* p0 = x + (size_t)gy * WW + gx;
      float xv[CC];
      float mx = -3.402823466e38f;
#pragma unroll
      for (int c2 = 0; c2 < CC; ++c2) {
        xv[c2] = p0[(size_t)c2 * HH * WW];
        mx = fmaxf(mx, xv[c2]);
      }
      float s = 0.f;
#pragma unroll
      for (int c2 = 0; c2 < CC; ++c2) s += __expf(xv[c2] - mx);
      m = mx;
      r = 1.0f / s;
    }
    sm_m[hp] = m;
    sm_r[hp] = r; // r==0 marks out-of-image halo pixels
  }
  __syncthreads();

  const int tileId = ty * NTX + tx;
  for (int c = 0; c < CC; ++c) {
    const int buf = c & 1;
#if HAVE_TDM
    if (!edge) {
      if (tid < 32) { // wave 0: prefetch next channel tile, wait for current one
        if (c + 1 < CC) {
          tdm_load_tile(x + (size_t)(c + 1) * HH * WW + (size_t)gy0 * WW + gx0,
                        &sm_x[buf ^ 1][0]);
          tdm_wait(1);
        } else {
          tdm_wait(0);
        }
      }
    } else
#endif
    {
      for (int hp = tid; hp < HPIX; hp += 256) {
        int hy = hp / HALO, hx = hp - hy * HALO;
        int gy = gy0 + hy, gx = gx0 + hx;
        float v = 0.f;
        if ((unsigned)gy < (unsigned)HH && (unsigned)gx < (unsigned)WW)
          v = x[(size_t)c * HH * WW + (size_t)gy * WW + gx];
        sm_x[buf][hp] = v;
      }
    }
    __syncthreads();

    // p = softmax prob of this channel; -inf outside the image (matches -inf padding).
    for (int hp = tid; hp < HPIX; hp += 256) {
      float r = sm_r[hp];
      sm_p[hp] = (r > 0.f) ? __expf(sm_x[buf][hp] - sm_m[hp]) * r : -__builtin_inff();
    }
    __syncthreads();

    // Per-thread stats over 4 interior pixels.
    float lmin = 3.402823466e38f, ls = 0.f, lt1 = 0.f, lt2 = 0.f;
    int lc = 0;
#pragma unroll
    for (int rr = 0; rr < 4; ++rr) {
      int row = (tid >> 5) + rr * 8;
      int col = tid & 31;
      int hp = (row + 1) * HALO + (col + 1);
      float pc = sm_p[hp];
      lmin = fminf(lmin, pc);
      float nb = sm_p[hp - 1];
      nb = fmaxf(nb, sm_p[hp + 1]);
      nb = fmaxf(nb, sm_p[hp - HALO - 1]);
      nb = fmaxf(nb, sm_p[hp - HALO]);
      nb = fmaxf(nb, sm_p[hp - HALO + 1]);
      nb = fmaxf(nb, sm_p[hp + HALO - 1]);
      nb = fmaxf(nb, sm_p[hp + HALO]);
      nb = fmaxf(nb, sm_p[hp + HALO + 1]);
      if (pc > nb) { // strict 8-neighbor local maximum
        ls += pc;
        lc += 1;
        if (pc > lt1) { lt2 = lt1; lt1 = pc; }
        else if (pc > lt2) lt2 = pc;
      }
    }

    // Fixed-order workgroup tree reduction (deterministic).
    rS[tid] = ls; rM[tid] = lmin; rA[tid] = lt1; rB[tid] = lt2; rI[tid] = lc;
    __syncthreads();
    for (int s2 = 128; s2 > 0; s2 >>= 1) {
      if (tid < s2) {
        int o = tid + s2;
        rS[tid] += rS[o];
        rM[tid] = fminf(rM[tid], rM[o]);
        float a1 = rA[tid], a2 = rB[tid], b1 = rA[o], b2 = rB[o];
        rA[tid] = fmaxf(a1, b1);
        rB[tid] = fmaxf(fminf(a1, b1), fmaxf(a2, b2));
        rI[tid] += rI[o];
      }
      __syncthreads();
    }

    if (tid == 0) {
      psum[c * NTILES + tileId] = rS[0]; // non-atomic partial sum slot
      atomicAdd(&hdr[c], (unsigned int)rI[0]);
      atomicMin(&hdr[CC + c], __float_as_uint(rM[0])); // p >= 0: uint order == float order
      // Lock-free global top-2 via atomicMax displacement (value-deterministic).
      float tv[2] = { rA[0], rB[0] };
#pragma unroll
      for (int j = 0; j < 2; ++j) {
        if (tv[j] > 0.f) {
          unsigned int v = __float_as_uint(tv[j]);
          unsigned int old = atomicMax(&hdr[2 * CC + c], v);
          unsigned int cand = (v > old) ? old : v;
          atomicMax(&hdr[3 * CC + c], cand);
        }
      }
    }
    __syncthreads();
  }
}

// One wave; EXEC all-ones everywhere WMMA executes. Reduces 4096 partials/channel
// with v_wmma_f32_16x16x4_f32 (A = ones => sum of all D entries = 16 * sum(B entries),
// independent of the exact VGPR layout), then assembles the loss.
__global__ __launch_bounds__(32) void k_final(const unsigned int* __restrict__ hdr,
                                              const float* __restrict__ psum,
                                              float* __restrict__ out) {
  __shared__ float sred[32];
  const int lane = threadIdx.x;
  const v2f ones = { 1.0f, 1.0f };
  float lsum = 0.f;
  for (int c = 0; c < CC; ++c) {
    v8f acc = {};
    for (int base = 0; base < NTILES; base += 64) {
      int idx = c * NTILES + base + 2 * lane;
      v2f b;
      b.x = psum[idx];
      b.y = psum[idx + 1];
      acc = __builtin_amdgcn_wmma_f32_16x16x4_f32(false, ones, false, b,
                                                  (short)0, acc, false, false);
    }
    float part = acc[0] + acc[1] + acc[2] + acc[3] + acc[4] + acc[5] + acc[6] + acc[7];
    sred[lane] = part;
    __syncthreads();
    float S = 0.f;
    for (int i = 0; i < 32; ++i) S += sred[i]; // fixed order, all lanes identical
    S *= (1.0f / 16.0f);
    __syncthreads();

    unsigned int cnt = hdr[c];
    float m  = __uint_as_float(hdr[CC + c]);
    float t1 = __uint_as_float(hdr[2 * CC + c]);
    float t2 = __uint_as_float(hdr[3 * CC + c]);
    int k = (c < 7) ? 1 : 2;              // NUM_TARGET
    int nv = ((int)cnt < k) ? (int)cnt : k;
    float total = S - (float)cnt * m;     // sum of imp over all maxima
    float target = 0.f, hinge = 0.f;
    if (nv >= 1) { float v1 = t1 - m; target += v1; hinge += fmaxf(1.0f - v1, 0.f); }
    if (nv >= 2) { float v2 = t2 - m; target += v2; hinge += fmaxf(1.0f - v2, 0.f); }
    lsum += (total - target) + hinge;
  }
  if (lane == 0) out[0] = lsum * (1.0f / 14.0f);
}

extern "C" void kernel_launch(void* const* d_in, const int* in_sizes, int n_in,
                              void* d_out, int out_size, void* d_ws, size_t ws_size,
                              hipStream_t stream) {
  (void)in_sizes; (void)n_in; (void)out_size; (void)ws_size;
  const float* x = (const float*)d_in[0]; // [1,14,2048,2048] f32 logits
  // d_in[1] (int64 target) is unused by the reference loss value.
  unsigned int* hdr = (unsigned int*)d_ws;                 // 4*14 uints
  float* psum = (float*)((char*)d_ws + 256);               // 14*4096 f32 partial sums
  float* out = (float*)d_out;

  k_init<<<1, 64, 0, stream>>>(hdr);
  k_tile<<<dim3(NTX, NTY), 256, 0, stream>>>(x, hdr, psum);
  k_final<<<1, 32, 0, stream>>>(hdr, psum, out);
}